// FlyingInvPen_dynamics_63660005261426
// MI455X (gfx1250) — compile-verified
//
#include <hip/hip_runtime.h>

typedef __attribute__((ext_vector_type(4))) float v4f;

namespace {
constexpr float INVM = 1.0f / 2.1f;          // 1/MASS
constexpr float KF   = 0.0244101f;
constexpr float KM   = 0.00029958f;
constexpr float BF   = -30.48576f;
constexpr float BFZ4 = 4.0f * (-30.48576f);  // BF_VEC.z
constexpr float GZ   = -9.81f;
constexpr float MGZ  = 2.1f * (-9.81f);      // MASS * g_z
constexpr float L2   = 0.25f;                // L*L
constexpr float DT   = 0.05f;
constexpr float HDT  = 0.025f;               // 0.5*DT
constexpr float DT6  = 0.05f / 6.0f;

// J (block diagonal) and its inverse, folded at compile time in double.
constexpr double Ja = 0.01566089, Jb = 3.18037e-06, Jc = 0.01562078, Jd = 0.02226868;
constexpr double Jdet = Ja * Jc - Jb * Jb;
constexpr float JI00 = (float)( Jc / Jdet);
constexpr float JI01 = (float)(-Jb / Jdet);
constexpr float JI11 = (float)( Ja / Jdet);
constexpr float JI22 = (float)( 1.0 / Jd);
constexpr float JA = (float)Ja, JB = (float)Jb, JC = (float)Jc, JD = (float)Jd;

// MOTOR_DIST * (1/sqrt(2))  (SS row components)
constexpr float MDS = (float)(0.28 * 0.70710678118654752440);
} // namespace

struct V3 { float x, y, z; };

__device__ __forceinline__ V3 crossv(V3 a, V3 b) {
    return { a.y*b.z - a.z*b.y, a.z*b.x - a.x*b.z, a.x*b.y - a.y*b.x };
}

// q = (qw, qv); rotate v:  v + qw*t + qv x t,  t = 2*(qv x v)
__device__ __forceinline__ V3 quatrot(float qw, V3 qv, V3 v) {
    V3 t = crossv(qv, v);
    t.x *= 2.0f; t.y *= 2.0f; t.z *= 2.0f;
    V3 c = crossv(qv, t);
    return { v.x + qw*t.x + c.x, v.y + qw*t.y + c.y, v.z + qw*t.z + c.z };
}

// k = dynamics(s, u)  with u already scaled by ACT_SCALE
__device__ __forceinline__ void dyn(const float s[16], const float u[4], float k[16]) {
    const V3 m = { s[3],  s[4],  s[5]  };
    const V3 v = { s[6],  s[7],  s[8]  };
    const V3 w = { s[9],  s[10], s[11] };
    const float aa = s[12], bb = s[13];
    const float ad = s[14], bd = s[15];

    // MRP -> quaternion (and inverse)
    const float n2   = m.x*m.x + m.y*m.y + m.z*m.z;
    const float invs = 1.0f / (1.0f + n2);
    const float qw   = (1.0f - n2) * invs;
    const V3 qv  = {  2.0f*m.x*invs,  2.0f*m.y*invs,  2.0f*m.z*invs };
    const V3 qvn = { -qv.x, -qv.y, -qv.z };

    // body-frame force: thrust + rotated gravity + BF_VEC (drag term is 0: CD==0)
    const float Fz = KF * (u[0] + u[1] + u[2] + u[3]);
    V3 f = quatrot(qw, qvn, V3{0.0f, 0.0f, MGZ});
    f.z = Fz + f.z + BFZ4;

    // torque: yaw drag + motor-arm cross terms (summed in motor order 0..3)
    const float t0 = KF*u[0] + BF, t1 = KF*u[1] + BF;
    const float t2 = KF*u[2] + BF, t3 = KF*u[3] + BF;
    V3 tq;
    tq.x = ((MDS*t0 - MDS*t1) - MDS*t2) + MDS*t3;
    tq.y = ((-MDS*t0 - MDS*t1) + MDS*t2) + MDS*t3;
    tq.z = KM * (u[0] - u[1] + u[2] - u[3]);

    // mdot
    const float mdw = m.x*w.x + m.y*w.y + m.z*w.z;
    const V3 mxw = crossv(m, w);
    V3 mdot;
    mdot.x = 0.25f * ((1.0f - n2)*w.x + 2.0f*mxw.x + 2.0f*m.x*mdw);
    mdot.y = 0.25f * ((1.0f - n2)*w.y + 2.0f*mxw.y + 2.0f*m.y*mdw);
    mdot.z = 0.25f * ((1.0f - n2)*w.z + 2.0f*mxw.z + 2.0f*m.z*mdw);

    // pdot, vdot
    const V3 pdot = quatrot(qw, qv, v);
    const V3 wxv  = crossv(w, v);
    const V3 vdot = { f.x*INVM - wxv.x, f.y*INVM - wxv.y, f.z*INVM - wxv.z };

    // wdot = JINV @ (torque - w x (J w))
    const V3 Jw  = { JA*w.x + JB*w.y, JB*w.x + JC*w.y, JD*w.z };
    const V3 wJw = crossv(w, Jw);
    const V3 tm  = { tq.x - wJw.x, tq.y - wJw.y, tq.z - wJw.z };
    const V3 wdot = { JI00*tm.x + JI01*tm.y, JI01*tm.x + JI11*tm.y, JI22*tm.z };

    // world-frame acceleration
    const V3 vW = quatrot(qw, qv, vdot);

    // pendulum accelerations
    const float aa2 = aa*aa, bb2 = bb*bb;
    const float psi2 = L2 - aa2 - bb2;
    const float psi  = sqrtf(psi2);
    const float H = 4.0f*bd*bd*(aa2 - L2) - 8.0f*ad*bd*aa*bb
                  + 4.0f*ad*ad*(bb2 - L2) + 3.0f*psi*psi2*GZ;
    const float Hd = H / (4.0f*L2*psi2);
    const float inv4L2 = 1.0f / (4.0f*L2);
    const float B00 = 3.0f*(aa2 - L2)*inv4L2, B01 = 3.0f*aa*bb*inv4L2, B02 = 3.0f*aa*psi*inv4L2;
    const float B11 = 3.0f*(bb2 - L2)*inv4L2, B12 = 3.0f*bb*psi*inv4L2;
    const float add0 = aa*Hd + (B00*vW.x + B01*vW.y + B02*vW.z);
    const float add1 = bb*Hd + (B01*vW.x + B11*vW.y + B12*vW.z);

    k[0]  = pdot.x; k[1]  = pdot.y; k[2]  = pdot.z;
    k[3]  = mdot.x; k[4]  = mdot.y; k[5]  = mdot.z;
    k[6]  = vdot.x; k[7]  = vdot.y; k[8]  = vdot.z;
    k[9]  = wdot.x; k[10] = wdot.y; k[11] = wdot.z;
    k[12] = ad;     k[13] = bd;     k[14] = add0;   k[15] = add1;
}

__global__ __launch_bounds__(256) void flying_invpen_rk4(const float* __restrict__ x,
                                                         const float* __restrict__ u,
                                                         float* __restrict__ out,
                                                         int n) {
    const int i = blockIdx.x * blockDim.x + threadIdx.x;
    if (i >= n) return;

    // streaming loads: rows never re-read, dataset >> L2 -> non-temporal
    const v4f* x4 = reinterpret_cast<const v4f*>(x) + (size_t)i * 4;
    const v4f r0 = __builtin_nontemporal_load(x4 + 0);
    const v4f r1 = __builtin_nontemporal_load(x4 + 1);
    const v4f r2 = __builtin_nontemporal_load(x4 + 2);
    const v4f r3 = __builtin_nontemporal_load(x4 + 3);
    const v4f uu = __builtin_nontemporal_load(reinterpret_cast<const v4f*>(u) + i);

    float s0[16];
    #pragma unroll
    for (int j = 0; j < 4; ++j) {
        s0[j] = r0[j]; s0[4 + j] = r1[j]; s0[8 + j] = r2[j]; s0[12 + j] = r3[j];
    }
    float us[4];
    #pragma unroll
    for (int j = 0; j < 4; ++j) us[j] = 100.0f * uu[j];   // ACT_SCALE

    float k[16], acc[16], st[16];

    dyn(s0, us, k);                                       // k1
    #pragma unroll
    for (int j = 0; j < 16; ++j) { acc[j] = k[j]; st[j] = s0[j] + HDT*k[j]; }

    dyn(st, us, k);                                       // k2
    #pragma unroll
    for (int j = 0; j < 16; ++j) { acc[j] += 2.0f*k[j]; st[j] = s0[j] + HDT*k[j]; }

    dyn(st, us, k);                                       // k3
    #pragma unroll
    for (int j = 0; j < 16; ++j) { acc[j] += 2.0f*k[j]; st[j] = s0[j] + DT*k[j]; }

    dyn(st, us, k);                                       // k4

    v4f o0, o1, o2, o3;
    #pragma unroll
    for (int j = 0; j < 4; ++j) {
        o0[j] = s0[j]      + DT6 * (acc[j]      + k[j]);
        o1[j] = s0[4 + j]  + DT6 * (acc[4 + j]  + k[4 + j]);
        o2[j] = s0[8 + j]  + DT6 * (acc[8 + j]  + k[8 + j]);
        o3[j] = s0[12 + j] + DT6 * (acc[12 + j] + k[12 + j]);
    }
    v4f* out4 = reinterpret_cast<v4f*>(out) + (size_t)i * 4;
    __builtin_nontemporal_store(o0, out4 + 0);
    __builtin_nontemporal_store(o1, out4 + 1);
    __builtin_nontemporal_store(o2, out4 + 2);
    __builtin_nontemporal_store(o3, out4 + 3);
}

extern "C" void kernel_launch(void* const* d_in, const int* in_sizes, int n_in,
                              void* d_out, int out_size, void* d_ws, size_t ws_size,
                              hipStream_t stream) {
    const float* x = (const float*)d_in[0];   // (BSZ, 16) fp32
    const float* u = (const float*)d_in[1];   // (BSZ, 4)  fp32
    float* out = (float*)d_out;               // (BSZ, 16) fp32

    const int n = in_sizes[0] / 16;
    const int threads = 256;
    const int blocks = (n + threads - 1) / threads;
    flying_invpen_rk4<<<blocks, threads, 0, stream>>>(x, u, out, n);
}